// AttentiveFPReadout_25409026524087
// MI455X (gfx1250) — compile-verified
//
#include <hip/hip_runtime.h>
#include <hip/hip_bf16.h>
#include <math.h>

typedef __attribute__((ext_vector_type(2))) float v2f;
typedef __attribute__((ext_vector_type(8))) float v8f;

constexpr int FD = 256;     // feature dim
constexpr int FD3 = 768;    // 3*F
constexpr int LSTRIDE = 260; // LDS row stride (dwords): 16B-aligned rows, conflict-free b64 reads

// ---------------------------------------------------------------- utilities
__device__ __forceinline__ float wave_sum32(float v) {
#pragma unroll
  for (int off = 16; off > 0; off >>= 1) v += __shfl_xor(v, off, 32);
  return v;
}

__device__ __forceinline__ void async_copy_b128(unsigned lds_byte_off,
                                                const float* gsrc) {
  unsigned long long ga = (unsigned long long)(uintptr_t)gsrc;
  asm volatile("global_load_async_to_lds_b128 %0, %1, off"
               :: "v"(lds_byte_off), "v"(ga) : "memory");
}

__device__ __forceinline__ void async_wait0() {
  asm volatile("s_wait_asynccnt 0x0" ::: "memory");
}

// ------------------------------------------------- segment start offsets
__global__ void k_starts(const int* __restrict__ seg, int* __restrict__ start,
                         int V, int G) {
  int v = blockIdx.x * blockDim.x + threadIdx.x;
  if (v < V) {
    int s = seg[v];
    int p = (v == 0) ? -1 : seg[v - 1];
    for (int g = p + 1; g <= s; ++g) start[g] = v;
  } else if (v == V) {
    int last = seg[V - 1];
    for (int g = last + 1; g <= G; ++g) start[g] = V;
  }
}

// ------------------------------------------------- initial g_feats = segsum
__global__ void k_gsum(const float* __restrict__ x, const int* __restrict__ start,
                       float* __restrict__ gf) {
  int g = blockIdx.x, f = threadIdx.x;
  int s0 = start[g], s1 = start[g + 1];
  float acc = 0.f;
  for (int v = s0; v < s1; ++v) acc += x[(size_t)v * FD + f];
  gf[(size_t)g * FD + f] = acc;
}

// ------------------------------------------------- per-graph attention dot
__global__ void k_gdot(const float* __restrict__ gf, const float* __restrict__ Wl_t,
                       const float* __restrict__ bl_t, float* __restrict__ gdot) {
  __shared__ float red[256];
  int g = blockIdx.x, f = threadIdx.x;
  float v = gf[(size_t)g * FD + f];
  v = v > 0.f ? v : 0.f;                 // relu(g_feats)
  red[f] = v * Wl_t[f];
  __syncthreads();
#pragma unroll
  for (int s = 128; s > 0; s >>= 1) {
    if (f < s) red[f] += red[f + s];
    __syncthreads();
  }
  if (f == 0) gdot[g] = red[0] + bl_t[0];  // fold bias
}

// ------------------------------------------------- per-node logit z (leaky)
__global__ void k_z(const float* __restrict__ x, const int* __restrict__ seg,
                    const float* __restrict__ gdot, const float* __restrict__ Wl_t,
                    float* __restrict__ z, int V) {
  int lane = threadIdx.x & 31;
  int wid = threadIdx.x >> 5;
  int v = blockIdx.x * 8 + wid;
  if (v >= V) return;                      // wave-uniform exit
  const float4* x4 = (const float4*)(x + (size_t)v * FD);
  const float4* w4 = (const float4*)(Wl_t + FD);   // second half of Wl row
  float acc = 0.f;
#pragma unroll
  for (int i = 0; i < 2; ++i) {
    float4 a = x4[lane + 32 * i];
    float4 w = w4[lane + 32 * i];
    acc += a.x * w.x + a.y * w.y + a.z * w.z + a.w * w.w;
  }
  acc = wave_sum32(acc);
  if (lane == 0) {
    float s = acc + gdot[seg[v]];
    z[v] = s > 0.f ? s : 0.01f * s;        // leaky_relu
  }
}

// ------------------------- per-graph softmax stats + weighted feature sum
__global__ void k_soft(const float* __restrict__ x, const float* __restrict__ z,
                       const int* __restrict__ start, float* __restrict__ e,
                       float* __restrict__ wsum) {
  __shared__ float red[256];
  __shared__ float s_m, s_d;
  int g = blockIdx.x, f = threadIdx.x;
  int s0 = start[g], s1 = start[g + 1];
  size_t gbase = (size_t)g * FD;
  if (s0 >= s1) { wsum[gbase + f] = 0.f; return; }
  // max
  float m = -INFINITY;
  for (int v = s0 + f; v < s1; v += 256) m = fmaxf(m, z[v]);
  red[f] = m; __syncthreads();
#pragma unroll
  for (int s = 128; s > 0; s >>= 1) {
    if (f < s) red[f] = fmaxf(red[f], red[f + s]);
    __syncthreads();
  }
  if (f == 0) s_m = red[0];
  __syncthreads();
  m = s_m;
  // sum exp (stash e_v)
  float d = 0.f;
  for (int v = s0 + f; v < s1; v += 256) {
    float ev = __expf(z[v] - m);
    e[v] = ev;
    d += ev;
  }
  red[f] = d; __syncthreads();
#pragma unroll
  for (int s = 128; s > 0; s >>= 1) {
    if (f < s) red[f] += red[f + s];
    __syncthreads();
  }
  if (f == 0) s_d = red[0];
  __syncthreads();
  d = s_d;
  // weighted sum over nodes (coalesced across features)
  float acc = 0.f;
  for (int v = s0; v < s1; ++v) acc += e[v] * x[(size_t)v * FD + f];
  wsum[gbase + f] = acc / d;
}

// ----------------- WMMA: ctx = elu(wsum @ Wp.T + bp), empty graphs -> 0
// 256-thread block = 8 waves = 8 M-tiles; one Wp panel async-staged in LDS.
__global__ void k_ctx(const float* __restrict__ wsum, const float* __restrict__ Wp_t,
                      const float* __restrict__ bp_t, const int* __restrict__ start,
                      float* __restrict__ ctx, int G) {
  extern __shared__ float lds[];
  int tid = threadIdx.x;
  int lane = tid & 31, w = tid >> 5;
  int jt = blockIdx.y;
  int colbase = jt * 16;
  // async stage: Wp rows [colbase, colbase+16) -> lds[n*LSTRIDE + k]
  for (int c = tid; c < 16 * 64; c += 256) {
    int row = c >> 6, ch = c & 63;                 // ch = float4 chunk
    unsigned la = (unsigned)(uintptr_t)(lds + row * LSTRIDE + ch * 4);
    async_copy_b128(la, Wp_t + (size_t)(colbase + row) * FD + ch * 4);
  }
  async_wait0();
  __syncthreads();

  int mt = blockIdx.x * 8 + w;
  int half = lane >> 4, l16 = lane & 15;
  int arow = mt * 16 + l16;
  if (arow > G - 1) arow = G - 1;          // clamp: keep EXEC all-1s
  int koff = half * 2;
  const float* ap = wsum + (size_t)arow * FD + koff;
  const float* lb = lds + l16 * LSTRIDE + koff;
  v8f c = {0.f, 0.f, 0.f, 0.f, 0.f, 0.f, 0.f, 0.f};
#pragma unroll 4
  for (int k = 0; k < FD; k += 4) {
    v2f a = *(const v2f*)(ap + k);
    v2f b = *(const v2f*)(lb + k);
    c = __builtin_amdgcn_wmma_f32_16x16x4_f32(false, a, false, b, (short)0, c,
                                              false, false);
  }
  int col = colbase + l16;
  float bias = bp_t[col];
  int rbase = mt * 16 + half * 8;
#pragma unroll
  for (int r = 0; r < 8; ++r) {
    int row = rbase + r;
    if (row < G) {
      float val;
      if (start[row] == start[row + 1]) {
        val = 0.f;                          // empty segment: g_repr=0, elu(0)=0
      } else {
        float p = c[r] + bias;
        val = p > 0.f ? p : (__expf(p) - 1.f);  // elu
      }
      ctx[(size_t)row * FD + col] = val;
    }
  }
}

// ----------------- WMMA fused GRU cell: hout = GRU(ctx, h)
// 8 waves/block share six async-staged weight panels (r/z/n x ih/hh) in LDS.
__global__ void k_gru(const float* __restrict__ ctx, const float* __restrict__ h,
                      const float* __restrict__ Wih_t, const float* __restrict__ Whh_t,
                      const float* __restrict__ bih_t, const float* __restrict__ bhh_t,
                      float* __restrict__ hout, int G) {
  extern __shared__ float lds[];
  int tid = threadIdx.x;
  int lane = tid & 31, w = tid >> 5;
  int jt = blockIdx.y;
  int colbase = jt * 16;
  const int PANEL = 16 * LSTRIDE;
#pragma unroll
  for (int p = 0; p < 6; ++p) {
    const float* base = (p < 3) ? Wih_t + (size_t)(p * FD + colbase) * FD
                                : Whh_t + (size_t)((p - 3) * FD + colbase) * FD;
    for (int c = tid; c < 16 * 64; c += 256) {
      int row = c >> 6, ch = c & 63;
      unsigned la = (unsigned)(uintptr_t)(lds + p * PANEL + row * LSTRIDE + ch * 4);
      async_copy_b128(la, base + (size_t)row * FD + ch * 4);
    }
  }
  async_wait0();
  __syncthreads();

  int mt = blockIdx.x * 8 + w;
  int half = lane >> 4, l16 = lane & 15;
  int arow = mt * 16 + l16;
  if (arow > G - 1) arow = G - 1;
  int koff = half * 2;
  const float* ax = ctx + (size_t)arow * FD + koff;
  const float* ah = h + (size_t)arow * FD + koff;
  const float* lb0 = lds + 0 * PANEL + l16 * LSTRIDE + koff;
  const float* lb1 = lds + 1 * PANEL + l16 * LSTRIDE + koff;
  const float* lb2 = lds + 2 * PANEL + l16 * LSTRIDE + koff;
  const float* lb3 = lds + 3 * PANEL + l16 * LSTRIDE + koff;
  const float* lb4 = lds + 4 * PANEL + l16 * LSTRIDE + koff;
  const float* lb5 = lds + 5 * PANEL + l16 * LSTRIDE + koff;
  v8f zv = {0.f, 0.f, 0.f, 0.f, 0.f, 0.f, 0.f, 0.f};
  v8f cir = zv, ciz = zv, cin = zv, dhr = zv, dhz = zv, dhn = zv;
#pragma unroll 2
  for (int k = 0; k < FD; k += 4) {
    v2f a0 = *(const v2f*)(ax + k);
    v2f a1 = *(const v2f*)(ah + k);
    v2f b;
    b = *(const v2f*)(lb0 + k);
    cir = __builtin_amdgcn_wmma_f32_16x16x4_f32(false, a0, false, b, (short)0, cir, false, false);
    b = *(const v2f*)(lb1 + k);
    ciz = __builtin_amdgcn_wmma_f32_16x16x4_f32(false, a0, false, b, (short)0, ciz, false, false);
    b = *(const v2f*)(lb2 + k);
    cin = __builtin_amdgcn_wmma_f32_16x16x4_f32(false, a0, false, b, (short)0, cin, false, false);
    b = *(const v2f*)(lb3 + k);
    dhr = __builtin_amdgcn_wmma_f32_16x16x4_f32(false, a1, false, b, (short)0, dhr, false, false);
    b = *(const v2f*)(lb4 + k);
    dhz = __builtin_amdgcn_wmma_f32_16x16x4_f32(false, a1, false, b, (short)0, dhz, false, false);
    b = *(const v2f*)(lb5 + k);
    dhn = __builtin_amdgcn_wmma_f32_16x16x4_f32(false, a1, false, b, (short)0, dhn, false, false);
  }
  int col = colbase + l16;
  float b_ir = bih_t[col], b_iz = bih_t[FD + col], b_in = bih_t[2 * FD + col];
  float b_hr = bhh_t[col], b_hz = bhh_t[FD + col], b_hn = bhh_t[2 * FD + col];
  int rbase = mt * 16 + half * 8;
#pragma unroll
  for (int r = 0; r < 8; ++r) {
    int row = rbase + r;
    if (row < G) {
      float rg = 1.f / (1.f + __expf(-((cir[r] + b_ir) + (dhr[r] + b_hr))));
      float zg = 1.f / (1.f + __expf(-((ciz[r] + b_iz) + (dhz[r] + b_hz))));
      float nn = tanhf((cin[r] + b_in) + rg * (dhn[r] + b_hn));
      float ho = h[(size_t)row * FD + col];
      hout[(size_t)row * FD + col] = (1.f - zg) * nn + zg * ho;
    }
  }
}

// ---------------------------------------------------------------- launcher
extern "C" void kernel_launch(void* const* d_in, const int* in_sizes, int n_in,
                              void* d_out, int out_size, void* d_ws, size_t ws_size,
                              hipStream_t stream) {
  const float* x   = (const float*)d_in[0];
  const int*   seg = (const int*)d_in[1];
  const float* Wl  = (const float*)d_in[2];
  const float* bl  = (const float*)d_in[3];
  const float* Wp  = (const float*)d_in[4];
  const float* bp  = (const float*)d_in[5];
  const float* Wih = (const float*)d_in[6];
  const float* Whh = (const float*)d_in[7];
  const float* bih = (const float*)d_in[8];
  const float* bhh = (const float*)d_in[9];
  const int V = in_sizes[1];
  const int G = out_size / FD;
  const int T = in_sizes[3];           // bl has shape (T,1)

  char* ws = (char*)d_ws;
  auto carve = [&](size_t bytes) -> char* {
    char* p = ws;
    ws += (bytes + 255) & ~(size_t)255;
    return p;
  };
  int*   start = (int*)  carve(sizeof(int) * (size_t)(G + 1));
  float* gdot  = (float*)carve(sizeof(float) * (size_t)G);
  float* z     = (float*)carve(sizeof(float) * (size_t)V);
  float* e     = (float*)carve(sizeof(float) * (size_t)V);
  float* wsum  = (float*)carve(sizeof(float) * (size_t)G * FD);
  float* ctx   = (float*)carve(sizeof(float) * (size_t)G * FD);
  float* hA    = (float*)carve(sizeof(float) * (size_t)G * FD);
  float* hB    = (float*)carve(sizeof(float) * (size_t)G * FD);

  k_starts<<<(V + 1 + 255) / 256, 256, 0, stream>>>(seg, start, V, G);
  k_gsum<<<G, 256, 0, stream>>>(x, start, hA);

  float* hcur = hA;
  int mblocks = (G + 127) / 128;                 // 8 M-tiles of 16 rows per block
  size_t lds_ctx = (size_t)16 * LSTRIDE * sizeof(float);        // 16.6 KB
  size_t lds_gru = (size_t)6 * 16 * LSTRIDE * sizeof(float);    // 97.5 KB
  for (int t = 0; t < T; ++t) {
    const float* Wl_t  = Wl + (size_t)t * 2 * FD;
    const float* bl_t  = bl + t;
    const float* Wp_t  = Wp + (size_t)t * FD * FD;
    const float* bp_t  = bp + (size_t)t * FD;
    const float* Wih_t = Wih + (size_t)t * FD3 * FD;
    const float* Whh_t = Whh + (size_t)t * FD3 * FD;
    const float* bih_t = bih + (size_t)t * FD3;
    const float* bhh_t = bhh + (size_t)t * FD3;
    float* hnext = (t == T - 1) ? (float*)d_out : ((hcur == hA) ? hB : hA);

    k_gdot<<<G, 256, 0, stream>>>(hcur, Wl_t, bl_t, gdot);
    k_z<<<(V + 7) / 8, 256, 0, stream>>>(x, seg, gdot, Wl_t, z, V);
    k_soft<<<G, 256, 0, stream>>>(x, z, start, e, wsum);
    k_ctx<<<dim3(mblocks, 16), 256, lds_ctx, stream>>>(wsum, Wp_t, bp_t, start,
                                                       ctx, G);
    k_gru<<<dim3(mblocks, 16), 256, lds_gru, stream>>>(ctx, hcur, Wih_t, Whh_t,
                                                       bih_t, bhh_t, hnext, G);
    hcur = hnext;
  }
}